// LocSE_64707977282331
// MI455X (gfx1250) — compile-verified
//
#include <hip/hip_runtime.h>

// LocSE for MI455X (gfx1250, wave32).
// B=4, N=65536, K=8, DIMS=2, F=32, C_RPPE=7.
// Flat row r in [0, B*N*K): center = r>>3 (== b*N+n), b = r>>19, k = r&7.
//
// One wave handles a 16-row tile:
//  - both half-waves compute the same 16 rows' rppe (matches the documented
//    V_WMMA_F32_16X16X4_F32 A-layout: lanes 0-15 carry K={0,1}, lanes 16-31 K={2,3})
//  - 7x32 MLP = 2 K-chunks x 2 column tiles = 4 x v_wmma_f32_16x16x4_f32
//  - W staged in LDS with an explicit zero row 7 -> all operand fetches are
//    unconditional ds_load_b32 with address selects (no exec-mask divergence)
//  - output is write-once streaming (512 MiB) -> nontemporal stores, keeping
//    the 32 MiB feats gather working set resident in the 192 MiB L2
//  - B-matrix (4x16) VGPR layout assumed transpose-symmetric to A:
//    V0 = rows {K0 | K2} across lane halves, V1 = {K1 | K3}. (Not tabulated
//    in the ISA doc; compile-only environment.)

typedef __attribute__((ext_vector_type(2))) float v2f;
typedef __attribute__((ext_vector_type(4))) float v4f;   // clang vector: OK for NT builtins
typedef __attribute__((ext_vector_type(8))) float v8f;

__global__ __launch_bounds__(256) void locse_wmma_kernel(
    const float* __restrict__ pc,     // (B,N,2)
    const float* __restrict__ feats,  // (B,N,32)
    const int*   __restrict__ nidx,   // (B,N,8) flat == row index space
    const float* __restrict__ W,      // (7,32) row-major
    const float* __restrict__ bias,   // (32,)
    float*       __restrict__ out)    // (B,N,8,64)
{
    constexpr int N = 65536;
    constexpr int F = 32;

    // LDS: 8x32 weight matrix (row 7 = zero pad) + 32 bias values.
    __shared__ float Wl[8 * F + F];   // 288 floats

    {
        const int t = threadIdx.x;
        Wl[t] = (t < 7 * F) ? W[t] : 0.0f;   // rows 0..6 = W, row 7 = 0
        if (t < F) Wl[8 * F + t] = bias[t];
    }
    __syncthreads();

    const int lane = threadIdx.x & 31;
    const int wave = threadIdx.x >> 5;
    const int tile = blockIdx.x * 8 + wave;     // wave-uniform, exact cover
    const int R0   = tile * 16;

    // ---------------- neighbor-feature copy (coalesced, 2 lanes / row) --------
    {
        const int row  = R0 + (lane >> 1);
        const int half = lane & 1;
        const int b_i  = row >> 19;                       // / (N*K)
        const int idx  = nidx[row];
        const v4f* src = reinterpret_cast<const v4f*>(
            feats + (size_t)(b_i * N + idx) * F + half * 16);
        v4f* dst = reinterpret_cast<v4f*>(
            out + (size_t)row * 64 + half * 16);
        __builtin_nontemporal_store(src[0], dst + 0);
        __builtin_nontemporal_store(src[1], dst + 1);
        __builtin_nontemporal_store(src[2], dst + 2);
        __builtin_nontemporal_store(src[3], dst + 3);
    }

    // ---------------- per-row rppe (each half-wave does the same 16 rows) -----
    const int  mrow   = R0 + (lane & 15);
    const int  center = mrow >> 3;            // b*N + n
    const int  b_i    = mrow >> 19;
    const int  idx    = nidx[mrow];
    const int  nb     = b_i * N + idx;

    const float cx = pc[center * 2 + 0];
    const float cy = pc[center * 2 + 1];
    const float nx = pc[nb * 2 + 0];
    const float ny = pc[nb * 2 + 1];
    const float rx = cx - nx;
    const float ry = cy - ny;
    const float nrm = __builtin_amdgcn_sqrtf(rx * rx + ry * ry);  // raw v_sqrt_f32

    const bool hi = lane >= 16;
    // rppe channels: 0:cx 1:cy 2:nx 3:ny 4:rx 5:ry 6:nrm 7:zero-pad
    v2f a0, a1;
    a0.x = hi ? nx  : cx;   // chunk0: K={0,1} (lanes 0-15) / K={2,3} (lanes 16-31)
    a0.y = hi ? ny  : cy;
    a1.x = hi ? nrm : rx;   // chunk1: K={4,5} / K={6,7=pad}
    a1.y = hi ? 0.f : ry;

    // Row indices into the LDS weight tile (zero row 7 makes these branch-free).
    const int r0a = (hi ? 2 : 0) * F;   // chunk0 V0
    const int r0b = (hi ? 3 : 1) * F;   // chunk0 V1
    const int r1a = (hi ? 6 : 4) * F;   // chunk1 V0
    const int r1b = (hi ? 7 : 5) * F;   // chunk1 V1 (row 7 == zeros)

    // ---------------- 7x32 MLP via f32 WMMA, two 16-column tiles --------------
    #pragma unroll
    for (int t = 0; t < 2; ++t) {
        const int col = t * 16 + (lane & 15);

        v2f b0, b1;
        b0.x = Wl[r0a + col];
        b0.y = Wl[r0b + col];
        b1.x = Wl[r1a + col];
        b1.y = Wl[r1b + col];

        const float bb = Wl[8 * F + col];
        v8f acc = {bb, bb, bb, bb, bb, bb, bb, bb};

        // D = A*B + C  (8 args: neg_a, A, neg_b, B, c_mod, C, reuse_a, reuse_b)
        acc = __builtin_amdgcn_wmma_f32_16x16x4_f32(
            false, a0, false, b0, (short)0, acc, false, false);
        acc = __builtin_amdgcn_wmma_f32_16x16x4_f32(
            false, a1, false, b1, (short)0, acc, false, false);

        // D layout: lanes 0-15 -> M = v, lanes 16-31 -> M = 8+v; N = lane&15.
        float* obase = out + (size_t)R0 * 64 + 32 + col + (hi ? 8 * 64 : 0);
        #pragma unroll
        for (int v = 0; v < 8; ++v) {
            const float r = acc[v] > 0.f ? acc[v] : 0.f;   // ReLU
            __builtin_nontemporal_store(r, obase + (size_t)v * 64);
        }
    }
}

extern "C" void kernel_launch(void* const* d_in, const int* in_sizes, int n_in,
                              void* d_out, int out_size, void* d_ws, size_t ws_size,
                              hipStream_t stream) {
    const float* pc    = (const float*)d_in[0];
    const float* feats = (const float*)d_in[1];
    const int*   nidx  = (const int*)d_in[2];   // JAX x64-disabled -> int32
    const float* W     = (const float*)d_in[3];
    const float* bias  = (const float*)d_in[4];
    float*       out   = (float*)d_out;

    constexpr int TOTAL_ROWS = 4 * 65536 * 8;       // 2,097,152
    constexpr int TILES      = TOTAL_ROWS / 16;     // 131,072 (one wave each)
    constexpr int WPB        = 8;                   // waves per 256-thread block
    dim3 block(256);
    dim3 grid(TILES / WPB);                         // 16,384 blocks, exact cover

    locse_wmma_kernel<<<grid, block, 0, stream>>>(pc, feats, nidx, W, bias, out);
}